// Track_21543555957439
// MI455X (gfx1250) — compile-verified
//
#include <hip/hip_runtime.h>
#include <hip/hip_bf16.h>

#define C_LIGHT 0.29979245f

struct V3 { float x, y, z; };

// db/dt = -sqrt(1-|b|^2) * (b x F) == sqrt(1-|b|^2) * (F x b)
__device__ __forceinline__ V3 db_dt(V3 b, V3 F) {
    float d = 1.0f - ((b.x * b.x + b.y * b.y) + b.z * b.z);
    float s = __builtin_amdgcn_sqrtf(d);   // single v_sqrt_f32: shortest-latency sqrt
    V3 o;
    o.x = s * (F.y * b.z - F.z * b.y);
    o.y = s * (F.z * b.x - F.x * b.z);
    o.z = s * (F.x * b.y - F.y * b.x);
    return o;
}

__global__ __launch_bounds__(32, 1)
void track_rk4_serial(const float* __restrict__ time,
                      const float* __restrict__ r0v,
                      const float* __restrict__ b0v,
                      const float* __restrict__ zb,     // n_elems+1 bounds
                      const float* __restrict__ fv,     // [n_elems][3]
                      float* __restrict__ out,          // [r (n*3) | beta (n*3)]
                      int n, int n_elems)
{
    if (threadIdx.x != 0) return;   // serial recurrence: one lane owns the chain

    // gfx1250 prefetch path (global_prefetch_b8) for the small lookup tables
    __builtin_prefetch(zb, 0, 0);
    __builtin_prefetch(fv, 0, 0);

    float* __restrict__ out_r = out;
    float* __restrict__ out_b = out + (size_t)3 * (size_t)n;

    const float dt  = time[1] - time[0];
    const float dt2 = dt * 0.5f;
    const float cr  = dt * C_LIGHT / 6.0f;
    const float cb  = dt / 6.0f;

    V3 r = { r0v[0], r0v[1], r0v[2] };
    V3 b = { b0v[0], b0v[1], b0v[2] };

    out_r[0] = r.x; out_r[1] = r.y; out_r[2] = r.z;
    out_b[0] = b.x; out_b[1] = b.y; out_b[2] = b.z;

    const float zb_first = zb[0];
    const float zb_last  = zb[n_elems];

    // Locate initial segment (matches searchsorted(z, 'right')-1 clipped to [0, n_elems-1])
    int idx = 0;
    while (idx < n_elems - 1 && r.z >= zb[idx + 1]) ++idx;
    while (idx > 0 && r.z < zb[idx]) --idx;
    float lo = zb[idx], hi = zb[idx + 1];
    bool inside = (r.z >= zb_first) && (r.z < zb_last);
    V3 F = { 0.0f, 0.0f, 0.0f };
    if (inside) { F.x = fv[3 * idx]; F.y = fv[3 * idx + 1]; F.z = fv[3 * idx + 2]; }

    for (int i = 1; i < n; ++i) {
        // RK4 with field held constant over the step (reference semantics)
        V3 k1 = db_dt(b, F);
        V3 b2 = { fmaf(dt2, k1.x, b.x), fmaf(dt2, k1.y, b.y), fmaf(dt2, k1.z, b.z) };
        V3 k2 = db_dt(b2, F);
        V3 b3 = { fmaf(dt2, k2.x, b.x), fmaf(dt2, k2.y, b.y), fmaf(dt2, k2.z, b.z) };
        V3 k3 = db_dt(b3, F);
        V3 b4 = { fmaf(dt, k3.x, b.x), fmaf(dt, k3.y, b.y), fmaf(dt, k3.z, b.z) };
        V3 k4 = db_dt(b4, F);

        // r_n = r + dt*C/6 * (r_k1 + 2 r_k2 + 2 r_k3 + r_k4), r_k1 = b, r_k2 = b2, ...
        r.x += cr * (((b.x + 2.0f * b2.x) + 2.0f * b3.x) + b4.x);
        r.y += cr * (((b.y + 2.0f * b2.y) + 2.0f * b3.y) + b4.y);
        r.z += cr * (((b.z + 2.0f * b2.z) + 2.0f * b3.z) + b4.z);
        // b_n = b + dt/6 * (k1 + 2 k2 + 2 k3 + k4)
        b.x += cb * (((k1.x + 2.0f * k2.x) + 2.0f * k3.x) + k4.x);
        b.y += cb * (((k1.y + 2.0f * k2.y) + 2.0f * k3.y) + k4.y);
        b.z += cb * (((k1.z + 2.0f * k2.z) + 2.0f * k3.z) + k4.z);

        // fire-and-forget stores; STOREcnt drains in the shadow of the VALU chain
        size_t o = (size_t)3 * (size_t)i;
        out_r[o + 0] = r.x; out_r[o + 1] = r.y; out_r[o + 2] = r.z;
        out_b[o + 0] = b.x; out_b[o + 1] = b.y; out_b[o + 2] = b.z;

        // Incremental O(1) segment tracking for next step's field query at new r.z.
        // Matches clip(searchsorted(zb, z, 'right')-1, 0, n_elems-1).
        bool moved = false;
        while (r.z >= hi && idx < n_elems - 1) { ++idx; lo = hi; hi = zb[idx + 1]; moved = true; }
        while (r.z <  lo && idx > 0)           { --idx; hi = lo; lo = zb[idx];     moved = true; }
        bool ins = (r.z >= zb_first) && (r.z < zb_last);
        if (moved || ins != inside) {
            inside = ins;
            if (inside) { F.x = fv[3 * idx]; F.y = fv[3 * idx + 1]; F.z = fv[3 * idx + 2]; }
            else        { F.x = 0.0f; F.y = 0.0f; F.z = 0.0f; }
        }
    }
}

extern "C" void kernel_launch(void* const* d_in, const int* in_sizes, int n_in,
                              void* d_out, int out_size, void* d_ws, size_t ws_size,
                              hipStream_t stream) {
    const float* time = (const float*)d_in[0];
    const float* r0   = (const float*)d_in[1];
    const float* b0   = (const float*)d_in[2];
    const float* zb   = (const float*)d_in[3];
    const float* fv   = (const float*)d_in[4];
    float* out = (float*)d_out;

    int n       = in_sizes[0];       // N_STEPS
    int n_elems = in_sizes[3] - 1;   // z_bounds has N_ELEMS+1 entries

    (void)d_ws; (void)ws_size; (void)n_in; (void)out_size;

    // One wave32: the recurrence is a single serial dependence chain.
    track_rk4_serial<<<1, 32, 0, stream>>>(time, r0, b0, zb, fv, out, n, n_elems);
}